// EventTrace_44753559224664
// MI455X (gfx1250) — compile-verified
//
#include <hip/hip_runtime.h>
#include <hip/hip_bf16.h>

typedef __attribute__((ext_vector_type(2))) float v2f;
typedef __attribute__((ext_vector_type(8))) float v8f;

#define ALPHA 0.9f
#define BB    16
#define TT    4096
#define DD    512
#define SEG   512   // T-segment per wave (8-way parallelism over T)
#define WARM  512   // warmup window: alpha^512 ~ 4e-24, below f32 result noise

// One wave (32 lanes) owns one (batch, 16-wide d-tile, T-segment).
// Chunk of 16 timesteps == D(16x16) = L(16x16 lower-tri alpha powers) x E(16x16)
// done as 4 chained V_WMMA_F32_16X16X4_F32; carry enters via alpha^(i+1) in the
// accumulator init and exits as row 15, broadcast with one wave32 shuffle.
// All global indices are kept in 32-bit so loads/stores lower to the
// saddr + 32-bit-voffset (+imm) form instead of per-element 64-bit adds.
__global__ __launch_bounds__(32)
void event_trace_wmma_kernel(const int* __restrict__ ctrl,
                             const float* __restrict__ prev,
                             const float* __restrict__ embed,
                             float* __restrict__ out) {
    const int lane = threadIdx.x;   // 0..31 (wave32)
    const int col  = lane & 15;     // N index (d within tile); also token row id
    const int half = lane >> 4;     // lane-half: selects K/M sub-ranges
    const int d0   = (int)blockIdx.x << 4;
    const int b    = (int)blockIdx.y;
    const int seg  = (int)blockIdx.z;
    const int d    = d0 + col;

    const int seg_start = seg * SEG;
    const int seg_end   = seg_start + SEG;
    const int t0        = (seg_start >= WARM) ? (seg_start - WARM) : 0;

    // alpha^k, k = 0..16 (registers, fully unrolled)
    float apow[17];
    apow[0] = 1.f;
#pragma unroll
    for (int k = 1; k <= 16; ++k) apow[k] = apow[k - 1] * ALPHA;

    // A fragments for the 4 K-groups of the 16x16 lower-triangular L.
    // f32 16x4 A layout: lanes 0-15: M=lane, K={j0, j0+1}; lanes 16-31: K={j0+2, j0+3}.
    v2f aK[4];
#pragma unroll
    for (int kg = 0; kg < 4; ++kg) {
        int j0 = kg * 4 + half * 2;
        aK[kg].x = (j0     <= col) ? apow[col - j0]     : 0.f;
        aK[kg].y = (j0 + 1 <= col) ? apow[col - j0 - 1] : 0.f;
    }

    // carry scale per accumulator slot: row i = v + 8*half -> alpha^(i+1)
    float cf[8];
#pragma unroll
    for (int v = 0; v < 8; ++v) cf[v] = apow[v + 8 * half + 1];

    // trace[t0 - 1]: exact prev_trace when history starts at t=0, else 0
    // (contributions older than WARM steps are below f32 significance).
    float carry = (t0 == 0) ? prev[b * DD + d] : 0.f;

    // 32-bit element indices (whole output is 32M elements < 2^31)
    const int tok_base = (b * TT + t0 + col) << 2;       // ctrl elem index, +64/chunk
    int tok_idx = tok_base + 1;                          // channel 1 = event type
    // out elem index for accumulator slot v at chunk tc: out_idx + v*DD
    unsigned out_idx = (unsigned)(b * TT + t0 + 8 * half) * DD + (unsigned)d;
    const unsigned out_step = 16u * DD;                  // advance per chunk

    for (int tc = t0; tc < seg_end; tc += 16) {
        // lanes 0..15 hold the event token for time-row `col` of this chunk
        int tok = ctrl[tok_idx];

        v8f acc;
#pragma unroll
        for (int v = 0; v < 8; ++v) acc[v] = cf[v] * carry;

#pragma unroll
        for (int kg = 0; kg < 4; ++kg) {
            int r0  = kg * 4 + half * 2;                 // B rows this lane supplies
            int tkA = __shfl(tok, r0, 32);
            int tkB = __shfl(tok, r0 + 1, 32);
            v2f bv;
            bv.x = embed[tkA * DD + d];                  // E[tc+r0  , d]
            bv.y = embed[tkB * DD + d];                  // E[tc+r0+1, d]
            acc = __builtin_amdgcn_wmma_f32_16x16x4_f32(
                      false, aK[kg], false, bv, (short)0, acc, false, false);
        }

        if (tc >= seg_start) {                           // skip stores during warmup
#pragma unroll
            for (int v = 0; v < 8; ++v)
                out[out_idx + (unsigned)(v * DD)] = acc[v];  // imm offset = v*2048B
        }

        // next carry = D[15][n], held by lane 16+n in acc slot 7
        carry = __shfl(acc[7], 16 + col, 32);

        tok_idx += 16 * 4;                               // 16 timesteps * 4 channels
        out_idx += out_step;
    }
}

extern "C" void kernel_launch(void* const* d_in, const int* in_sizes, int n_in,
                              void* d_out, int out_size, void* d_ws, size_t ws_size,
                              hipStream_t stream) {
    (void)in_sizes; (void)n_in; (void)out_size; (void)d_ws; (void)ws_size;
    const int*   ctrl  = (const int*)d_in[0];    // [B,T,4] int32
    const float* prev  = (const float*)d_in[1];  // [B,D] f32
    const float* embed = (const float*)d_in[2];  // [V,D] f32
    float*       out   = (float*)d_out;          // [B,T,D] f32

    dim3 grid(DD / 16, BB, TT / SEG);            // 32 x 16 x 8 = 4096 waves
    dim3 block(32);                              // one wave32 per block
    event_trace_wmma_kernel<<<grid, block, 0, stream>>>(ctrl, prev, embed, out);
}